// QMeasureClassif_32890859552776
// MI455X (gfx1250) — compile-verified
//
#include <hip/hip_runtime.h>

typedef float v2f __attribute__((ext_vector_type(2)));
typedef float v8f __attribute__((ext_vector_type(8)));
typedef int   v4i __attribute__((vector_size(16)));   // matches builtin's b128 type

#define AS1 __attribute__((address_space(1)))
#define AS3 __attribute__((address_space(3)))

#if __has_builtin(__builtin_amdgcn_global_load_async_to_lds_b32) && \
    __has_builtin(__builtin_amdgcn_global_load_async_to_lds_b128)
#define HAVE_ASYNC 1
#else
#define HAVE_ASYNC 0
#endif

#define BATCH 256
#define KDIM  1024
#define DY    10
#define NCOL  (KDIM * DY * DY)   // 102400 columns of rho_flat
#define TN    80                 // columns per workgroup (8 m's x 10 n's, one l)
#define NS    (TN / 16)          // 5 column subtiles per wave
#define KC    32                 // K-chunk staged in LDS
#define LDA   36                 // padded LDS row stride for x tile (floats)
#define NCH   (KDIM / KC)        // 32 chunks
#define NAB   (KC * TN / 4 / 256)   // 2: rho b32 asyncs per thread / ... (see below)

__device__ __forceinline__ void wait_async_le18() {
#if __has_builtin(__builtin_amdgcn_s_wait_asynccnt)
    __builtin_amdgcn_s_wait_asynccnt(18);
#else
    asm volatile("s_wait_asynccnt 0x12" ::: "memory");
#endif
}
__device__ __forceinline__ void wait_async_le0() {
#if __has_builtin(__builtin_amdgcn_s_wait_asynccnt)
    __builtin_amdgcn_s_wait_asynccnt(0);
#else
    asm volatile("s_wait_asynccnt 0x0" ::: "memory");
#endif
}

// rho tile in LDS is k-quad interleaved:
//   element (r, c) of a KCxTN chunk lives at ldsB[((r>>2)*TN + c)*4 + (r&3)]
// so a WMMA B fragment (k, k+1 at one column) is a single 8B ds_load_b64.

__global__ __launch_bounds__(256)
void qmc_gemm_kernel(const float* __restrict__ x,
                     const float* __restrict__ rho,
                     float* __restrict__ Mbuf) {
#if HAVE_ASYNC
    __shared__ float ldsB[2][KC * TN];     // 2 x 10 KB rho ping-pong
    __shared__ float ldsA[2][BATCH * LDA]; // 2 x 36 KB x ping-pong
#else
    __shared__ float ldsB1[KC * TN];
    __shared__ float ldsA1[BATCH * LDA];
#endif
    __shared__ float ldsM[BATCH * DY];     // 10 KB per-WG partial M[b,n]

    const int tid  = threadIdx.x;
    const int lane = tid & 31;
    const int wave = tid >> 5;
    const int col0 = blockIdx.x * TN;
    const int lidx = col0 / (KDIM * DY);   // l constant across this tile

    for (int i = tid; i < BATCH * DY; i += 256) ldsM[i] = 0.0f;

    v8f acc[2][NS];                        // 80 accumulator VGPRs
#pragma unroll
    for (int t = 0; t < 2; ++t)
#pragma unroll
        for (int s = 0; s < NS; ++s)
            acc[t][s] = (v8f){0.f, 0.f, 0.f, 0.f, 0.f, 0.f, 0.f, 0.f};

    const int arow = lane & 15;            // M (or N) index within a 16-tile
    const int koff = (lane >> 4) << 1;     // lanes 16-31 hold K=2,3 of the quad
    const int M0   = wave * 32;
    const int M1   = M0 + 16;

    auto compute_chunk = [&](const float* __restrict__ aT,
                             const float* __restrict__ bT) {
        const float* aB0 = aT + (M0 + arow) * LDA + koff;
        const float* aB1 = aT + (M1 + arow) * LDA + koff;
        for (int kc = 0; kc < KC; kc += 4) {
            v2f a0 = *(const v2f*)(aB0 + kc);            // ds_load_b64
            v2f a1 = *(const v2f*)(aB1 + kc);            // ds_load_b64
            const float* bB = bT + (kc >> 2) * (TN * 4) + arow * 4 + koff;
#pragma unroll
            for (int s = 0; s < NS; ++s) {
                v2f b = *(const v2f*)(bB + s * 64);      // ds_load_b64
                acc[0][s] = __builtin_amdgcn_wmma_f32_16x16x4_f32(
                    false, a0, false, b, (short)0, acc[0][s], false, false);
                acc[1][s] = __builtin_amdgcn_wmma_f32_16x16x4_f32(
                    false, a1, false, b, (short)0, acc[1][s], false, false);
            }
        }
    };

#if HAVE_ASYNC
    // ---- async-to-LDS staging, ping-pong double buffer ----
    auto issue_chunk = [&](int kk0, int p) {
        AS3 float* bdst = (AS3 float*)(&ldsB[p][0]);
        AS3 float* adst = (AS3 float*)(&ldsA[p][0]);
#pragma unroll
        for (int j = 0; j < 10; ++j) {                   // KC*TN = 2560 b32
            const int i = tid + j * 256;
            const int r = i / TN, cc = i - r * TN;
            __builtin_amdgcn_global_load_async_to_lds_b32(
                (AS1 int*)(rho + (size_t)(kk0 + r) * NCOL + col0 + cc),
                (AS3 int*)(bdst + ((r >> 2) * TN + cc) * 4 + (r & 3)), 0, 0);
        }
#pragma unroll
        for (int j = 0; j < 8; ++j) {                    // BATCH*KC = 2048 f4
            const int i = tid + j * 256;
            const int r = i >> 3, c4 = i & 7;
            __builtin_amdgcn_global_load_async_to_lds_b128(
                (AS1 v4i*)(x + (size_t)r * KDIM + kk0 + c4 * 4),
                (AS3 v4i*)(adst + r * LDA + c4 * 4), 0, 0);
        }
    };

    issue_chunk(0, 0);
    for (int c = 0; c < NCH; ++c) {
        const int p = c & 1;
        __syncthreads();                   // all waves done reading buffer p^1
        if (c + 1 < NCH) {
            issue_chunk((c + 1) * KC, p ^ 1);
            wait_async_le18();             // in-order: chunk c fully resident
        } else {
            wait_async_le0();
        }
        __syncthreads();                   // chunk c visible to all waves
        compute_chunk(&ldsA[p][0], &ldsB[p][0]);
    }
#else
    // ---- fallback: register-staged double buffer ----
    v2f    rstage[5];                      // KC*TN = 2560 floats = 1280 v2f
    float4 xstage[8];
    auto load_chunk = [&](int kk0) {
#pragma unroll
        for (int j = 0; j < 5; ++j) {
            const int i = tid + j * 256;   // [0,1280) v2f index
            const int r = i / 40, c2 = i - r * 40;
            rstage[j] = *(const v2f*)(rho + (size_t)(kk0 + r) * NCOL + col0 + c2 * 2);
        }
#pragma unroll
        for (int j = 0; j < 8; ++j) {
            const int i = tid + j * 256;
            const int r = i >> 3, c4 = i & 7;
            xstage[j] = *(const float4*)(x + (size_t)r * KDIM + kk0 + c4 * 4);
        }
    };
    auto store_chunk = [&]() {
#pragma unroll
        for (int j = 0; j < 5; ++j) {
            const int i = tid + j * 256;
            const int r = i / 40, c2 = i - r * 40;
            const int base = ((r >> 2) * TN + c2 * 2) * 4 + (r & 3);
            ldsB1[base + 0] = rstage[j].x;
            ldsB1[base + 4] = rstage[j].y;
        }
#pragma unroll
        for (int j = 0; j < 8; ++j) {
            const int i = tid + j * 256;
            const int r = i >> 3, c4 = i & 7;
            *(float4*)(ldsA1 + r * LDA + c4 * 4) = xstage[j];
        }
    };
    load_chunk(0);
    for (int kk0 = 0; kk0 < KDIM; kk0 += KC) {
        __syncthreads();
        store_chunk();
        __syncthreads();
        if (kk0 + KC < KDIM) load_chunk(kk0 + KC);
        compute_chunk(ldsA1, ldsB1);
    }
#endif

    __syncthreads();

    // Epilogue: M[b,l,n] += T[b,col] * x[b,m(col)], reduced over m in LDS.
    // C/D layout: VGPR v -> row M = v + 8*(lane/16); lane%16 -> column.
    const int rowhi = (lane >> 4) * 8;
#pragma unroll
    for (int t = 0; t < 2; ++t) {
        const int mb = (t == 0) ? M0 : M1;
#pragma unroll
        for (int s = 0; s < NS; ++s) {
            const int col = col0 + s * 16 + arow;
            const int m   = (col % (KDIM * DY)) / DY;
            const int n   = col % DY;
#pragma unroll
            for (int v = 0; v < 8; ++v) {
                const int brow = mb + rowhi + v;
                const float val = acc[t][s][v] * x[(size_t)brow * KDIM + m];
                atomicAdd(&ldsM[brow * DY + n], val);    // ds_add_f32
            }
        }
    }
    __syncthreads();

    for (int i = tid; i < BATCH * DY; i += 256) {
        const int brow = i / DY;
        const int n    = i - brow * DY;
        atomicAdd(&Mbuf[brow * (DY * DY) + lidx * DY + n], ldsM[i]);
    }
}

__global__ void qmc_zero_kernel(float* __restrict__ p, int n) {
    const int i = blockIdx.x * blockDim.x + threadIdx.x;
    if (i < n) p[i] = 0.0f;
}

__global__ __launch_bounds__(128)
void qmc_norm_kernel(const float* __restrict__ Mbuf, float* __restrict__ out) {
    const int b = blockIdx.x;
    const int t = threadIdx.x;
    __shared__ float tr;
    if (t == 0) {
        float s = 0.0f;
#pragma unroll
        for (int d = 0; d < DY; ++d) s += Mbuf[b * (DY * DY) + d * (DY + 1)];
        tr = s;
    }
    __syncthreads();
    if (t < DY * DY) out[b * (DY * DY) + t] = Mbuf[b * (DY * DY) + t] / tr;
}

extern "C" void kernel_launch(void* const* d_in, const int* in_sizes, int n_in,
                              void* d_out, int out_size, void* d_ws, size_t ws_size,
                              hipStream_t stream) {
    const float* x   = (const float*)d_in[0];   // (256, 1024) fp32
    const float* rho = (const float*)d_in[1];   // (1024, 10, 1024, 10) fp32
    float* out  = (float*)d_out;                // (256, 10, 10) fp32
    float* Mbuf = (float*)d_ws;                 // 256*100 floats = 102.4 KB

    const int mcells = BATCH * DY * DY;
    qmc_zero_kernel<<<(mcells + 255) / 256, 256, 0, stream>>>(Mbuf, mcells);
    qmc_gemm_kernel<<<NCOL / TN, 256, 0, stream>>>(x, rho, Mbuf);
    qmc_norm_kernel<<<BATCH, 128, 0, stream>>>(Mbuf, out);
}